// KMeans_2980707304083
// MI455X (gfx1250) — compile-verified
//
#include <hip/hip_runtime.h>

#define D 64
#define KC 256
#define IRUNS 8
#define NITER 15

typedef float v2f __attribute__((ext_vector_type(2)));
typedef float v8f __attribute__((ext_vector_type(8)));

// ---------------------------------------------------------------------------
// Assignment: labels[i,n] = argmin_k ( c2[i,k] - 2 * x[n,:] . centers[i,k,:] )
// fp32 WMMA 16x16x4. Block = 8 waves = 128 points x 256 centers, one init run.
// ---------------------------------------------------------------------------
__global__ __launch_bounds__(256)
void kmeans_assign(const float* __restrict__ x,
                   const float* __restrict__ centers,   // (I,K,D)
                   const float* __restrict__ c2,        // (I,K)
                   int* __restrict__ labels,            // (I,N)
                   int N)
{
    __shared__ float cs[KC * D];                       // 64 KB center stage
    const int i = blockIdx.y;
    {
        const float4* s4 = (const float4*)(centers + (size_t)i * KC * D);
        float4* d4 = (float4*)cs;
        for (int t = threadIdx.x; t < KC * D / 4; t += blockDim.x) d4[t] = s4[t];
    }
    __syncthreads();

    const int wave = threadIdx.x >> 5;
    const int lane = threadIdx.x & 31;
    const int row  = lane & 15;        // M row for A-frag, N col for B-frag
    const int half = lane >> 4;

    const int m_base = blockIdx.x * 128 + wave * 16;
    const float* xrow = x + (size_t)(m_base + row) * D;

    // Preload this lane's A slices for all 16 k-steps (k = 4*ds + 2*half + {0,1})
    v2f a[16];
#pragma unroll
    for (int ds = 0; ds < 16; ++ds)
        a[ds] = *(const v2f*)(xrow + 4 * ds + 2 * half);

    const float* c2I = c2 + i * KC;

    float best[8];
    int   bidx[8];
#pragma unroll
    for (int r = 0; r < 8; ++r) { best[r] = 3.0e38f; bidx[r] = 0; }

    for (int ct = 0; ct < 16; ++ct) {                  // 16 center tiles of 16
        v8f acc = {0.f, 0.f, 0.f, 0.f, 0.f, 0.f, 0.f, 0.f};
        const float* crow = cs + (ct * 16 + row) * D;  // B[k][n]: n = row
#pragma unroll
        for (int ds = 0; ds < 16; ++ds) {
            v2f b = *(const v2f*)(crow + 4 * ds + 2 * half);
            acc = __builtin_amdgcn_wmma_f32_16x16x4_f32(
                false, a[ds], false, b, (short)0, acc, false, false);
        }
        const int   n   = ct * 16 + row;
        const float c2v = c2I[n];
#pragma unroll
        for (int r = 0; r < 8; ++r) {                  // point m = 8*half + r
            float dth = __builtin_fmaf(-2.0f, acc[r], c2v);
            if (dth < best[r]) { best[r] = dth; bidx[r] = n; }
        }
    }

    // argmin across the 16 lanes of each half (xor masks < 16 keep halves)
#pragma unroll
    for (int off = 8; off >= 1; off >>= 1) {
#pragma unroll
        for (int r = 0; r < 8; ++r) {
            float ob = __shfl_xor(best[r], off, 32);
            int   oi = __shfl_xor(bidx[r], off, 32);
            if (ob < best[r] || (ob == best[r] && oi < bidx[r])) {
                best[r] = ob; bidx[r] = oi;
            }
        }
    }
    if (row == 0) {
        int* lp = labels + (size_t)i * N + m_base + 8 * half;
#pragma unroll
        for (int r = 0; r < 8; ++r) lp[r] = bidx[r];
    }
}

// ---------------------------------------------------------------------------
// Zero the scatter accumulators.
// ---------------------------------------------------------------------------
__global__ void kmeans_zero(float* __restrict__ sums, float* __restrict__ cnts)
{
    int t = blockIdx.x * blockDim.x + threadIdx.x;
    if (t < IRUNS * KC * D) sums[t] = 0.0f;
    if (t < IRUNS * KC)     cnts[t] = 0.0f;
}

// ---------------------------------------------------------------------------
// Scatter-add: 64 threads per point (one per dim), coalesced atomics.
// ---------------------------------------------------------------------------
__global__ __launch_bounds__(256)
void kmeans_accum(const float* __restrict__ x, const int* __restrict__ labels,
                  float* __restrict__ sums, float* __restrict__ cnts, int N)
{
    const int i  = blockIdx.y;
    const int pt = blockIdx.x * 4 + (threadIdx.x >> 6);
    const int d  = threadIdx.x & 63;
    if (pt >= N) return;
    const int l = labels[(size_t)i * N + pt];
    atomicAdd(&sums[((size_t)i * KC + l) * D + d], x[(size_t)pt * D + d]);
    if (d == 0) atomicAdd(&cnts[i * KC + l], 1.0f);
}

// ---------------------------------------------------------------------------
// Center update: mean (empty clusters keep old center) + refresh c2.
// One block of 64 threads per (i,k).
// ---------------------------------------------------------------------------
__global__ __launch_bounds__(64)
void kmeans_update(const float* __restrict__ sums, const float* __restrict__ cnts,
                   float* __restrict__ centers, float* __restrict__ c2)
{
    const int ik = blockIdx.x;
    const int d  = threadIdx.x;
    const float cnt = cnts[ik];
    float cd = centers[(size_t)ik * D + d];
    if (cnt > 0.0f) cd = sums[(size_t)ik * D + d] / cnt;
    centers[(size_t)ik * D + d] = cd;

    float s = cd * cd;
#pragma unroll
    for (int off = 16; off >= 1; off >>= 1) s += __shfl_xor(s, off, 32);
    __shared__ float part[2];
    if ((d & 31) == 0) part[d >> 5] = s;
    __syncthreads();
    if (d == 0) c2[ik] = part[0] + part[1];
}

// ---------------------------------------------------------------------------
// Init: copy input centers into working buffer and compute c2 (inputs are
// never mutated).
// ---------------------------------------------------------------------------
__global__ __launch_bounds__(64)
void kmeans_setc(const float* __restrict__ src, float* __restrict__ centers,
                 float* __restrict__ c2)
{
    const int ik = blockIdx.x;
    const int d  = threadIdx.x;
    const float cd = src[(size_t)ik * D + d];
    centers[(size_t)ik * D + d] = cd;

    float s = cd * cd;
#pragma unroll
    for (int off = 16; off >= 1; off >>= 1) s += __shfl_xor(s, off, 32);
    __shared__ float part[2];
    if ((d & 31) == 0) part[d >> 5] = s;
    __syncthreads();
    if (d == 0) c2[ik] = part[0] + part[1];
}

// ---------------------------------------------------------------------------
extern "C" void kernel_launch(void* const* d_in, const int* in_sizes, int n_in,
                              void* d_out, int out_size, void* d_ws, size_t ws_size,
                              hipStream_t stream)
{
    const float* x  = (const float*)d_in[0];        // (N,D)
    const float* c0 = (const float*)d_in[1];        // (I,K,D)
    const int N = in_sizes[0] / D;                  // 65536

    float* out = (float*)d_out;                     // [centers | labels(int)]
    int*   labels_out = (int*)(out + (size_t)IRUNS * KC * D);

    // workspace layout
    float* centers = (float*)d_ws;                  // I*K*D
    float* c2      = centers + (size_t)IRUNS * KC * D;
    float* sums    = c2 + IRUNS * KC;               // I*K*D
    float* cnts    = sums + (size_t)IRUNS * KC * D; // I*K
    int*   labels  = (int*)(cnts + IRUNS * KC);     // I*N

    dim3 gAssign(N / 128, IRUNS);
    dim3 gAccum(N / 4, IRUNS);
    const int zeroBlocks = (IRUNS * KC * D + 255) / 256;

    kmeans_setc<<<IRUNS * KC, 64, 0, stream>>>(c0, centers, c2);

    for (int it = 0; it < NITER; ++it) {
        kmeans_assign<<<gAssign, 256, 0, stream>>>(x, centers, c2, labels, N);
        kmeans_zero<<<zeroBlocks, 256, 0, stream>>>(sums, cnts);
        kmeans_accum<<<gAccum, 256, 0, stream>>>(x, labels, sums, cnts, N);
        kmeans_update<<<IRUNS * KC, 64, 0, stream>>>(sums, cnts, centers, c2);
    }

    // final hard assignment straight into the output buffer
    kmeans_assign<<<gAssign, 256, 0, stream>>>(x, centers, c2, labels_out, N);
    hipMemcpyAsync(out, centers, sizeof(float) * (size_t)IRUNS * KC * D,
                   hipMemcpyDeviceToDevice, stream);
}